// GraphSAGE_62861141344333
// MI455X (gfx1250) — compile-verified
//
#include <hip/hip_runtime.h>
#include <hip/hip_bf16.h>

#define N_NODES 50000
#define DIM 128
#define NEDGE 800000
#define NTILES (N_NODES / 16)   // 3125, exact

typedef __attribute__((ext_vector_type(2))) float v2f;
typedef __attribute__((ext_vector_type(8))) float v8f;

// ---------------------------------------------------------------- utilities
__global__ void zero_f32(float* __restrict__ p, int n) {
    int i = blockIdx.x * blockDim.x + threadIdx.x;
    if (i < n) p[i] = 0.0f;
}

__global__ void count_edges(const int* __restrict__ dst, float* __restrict__ cnt, int e) {
    int i = blockIdx.x * blockDim.x + threadIdx.x;
    if (i < e) atomicAdd(&cnt[dst[i]], 1.0f);
}

// one wave (32 lanes) per edge: lane handles 4 consecutive features
__global__ __launch_bounds__(256)
void scatter_add(const float* __restrict__ feat, const int* __restrict__ src,
                 const int* __restrict__ dst, float* __restrict__ agg, int e) {
    int eid  = blockIdx.x * (blockDim.x >> 5) + (threadIdx.x >> 5);
    int lane = threadIdx.x & 31;
    if (eid >= e) return;
    int s = src[eid], d = dst[eid];
    float4 v = ((const float4*)(feat + (size_t)s * DIM))[lane];
    float* o = agg + (size_t)d * DIM + lane * 4;
    atomicAdd(o + 0, v.x);
    atomicAdd(o + 1, v.y);
    atomicAdd(o + 2, v.z);
    atomicAdd(o + 3, v.w);
}

// ------------------------------------------------- fused SAGE layer (WMMA)
// out[m][n] = relu( (agg[m]/max(cnt,1)) @ Wl + xin[m] @ Wr + b )
__global__ __launch_bounds__(256)
void sage_gemm_relu(const float* __restrict__ agg, const float* __restrict__ cnt,
                    const float* __restrict__ xin,
                    const float* __restrict__ Wl, const float* __restrict__ Wr,
                    const float* __restrict__ bias, float* __restrict__ out) {
    extern __shared__ float smem[];
    float* sWl = smem;                 // DIM*DIM
    float* sWr = smem + DIM * DIM;     // DIM*DIM
    float* sB  = smem + 2 * DIM * DIM; // DIM
    for (int i = threadIdx.x; i < DIM * DIM / 4; i += blockDim.x) {
        ((float4*)sWl)[i] = ((const float4*)Wl)[i];
        ((float4*)sWr)[i] = ((const float4*)Wr)[i];
    }
    if (threadIdx.x < DIM) sB[threadIdx.x] = bias[threadIdx.x];
    __syncthreads();

    int wave = threadIdx.x >> 5;
    int lane = threadIdx.x & 31;
    int tile = blockIdx.x * 8 + wave;
    if (tile >= NTILES) return;                 // wave-uniform exit: EXEC stays full
    int m0   = tile * 16;
    int row  = lane & 15;                       // A row / B,C column
    int koff = (lane >> 4) * 2;                 // K split across lane halves
    int m    = m0 + row;

    const float* aAgg = agg + (size_t)m * DIM;
    const float* aX   = xin + (size_t)m * DIM;
    float sc = 1.0f / fmaxf(cnt[m], 1.0f);      // mean normalization folded into A

    v8f acc[8];
    v8f zv = {0.f,0.f,0.f,0.f,0.f,0.f,0.f,0.f};
#pragma unroll
    for (int t = 0; t < 8; ++t) acc[t] = zv;

    for (int k = 0; k < DIM; k += 4) {
        v2f fA, fX;
        fA.x = aAgg[k + koff] * sc;  fA.y = aAgg[k + koff + 1] * sc;
        fX.x = aX[k + koff];         fX.y = aX[k + koff + 1];
#pragma unroll
        for (int t = 0; t < 8; ++t) {
            int n = t * 16 + row;
            v2f bL, bR;
            bL.x = sWl[(k + koff) * DIM + n];  bL.y = sWl[(k + koff + 1) * DIM + n];
            bR.x = sWr[(k + koff) * DIM + n];  bR.y = sWr[(k + koff + 1) * DIM + n];
            acc[t] = __builtin_amdgcn_wmma_f32_16x16x4_f32(false, fA, false, bL,
                                                           (short)0, acc[t], false, false);
            acc[t] = __builtin_amdgcn_wmma_f32_16x16x4_f32(false, fX, false, bR,
                                                           (short)0, acc[t], false, false);
        }
    }

    int half = lane >> 4;
#pragma unroll
    for (int t = 0; t < 8; ++t) {
        int n = t * 16 + row;
        float bv = sB[n];
#pragma unroll
        for (int v = 0; v < 8; ++v) {
            int mm = m0 + v + 8 * half;        // C layout: VGPR v -> rows v / v+8
            out[(size_t)mm * DIM + n] = fmaxf(acc[t][v] + bv, 0.0f);
        }
    }
}

// ------------------- fused SAGE layer 1 + regression head (h2 never stored)
__global__ __launch_bounds__(256)
void sage_gemm_head(const float* __restrict__ agg, const float* __restrict__ cnt,
                    const float* __restrict__ hin,
                    const float* __restrict__ Wl, const float* __restrict__ Wr,
                    const float* __restrict__ bias,
                    const float* __restrict__ Wfc, const float* __restrict__ bfc,
                    float* __restrict__ out) {
    extern __shared__ float smem[];
    float* sWl  = smem;
    float* sWr  = smem + DIM * DIM;
    float* sB   = smem + 2 * DIM * DIM;
    float* sWfc = sB + DIM;
    for (int i = threadIdx.x; i < DIM * DIM / 4; i += blockDim.x) {
        ((float4*)sWl)[i] = ((const float4*)Wl)[i];
        ((float4*)sWr)[i] = ((const float4*)Wr)[i];
    }
    if (threadIdx.x < DIM) { sB[threadIdx.x] = bias[threadIdx.x]; sWfc[threadIdx.x] = Wfc[threadIdx.x]; }
    __syncthreads();

    int wave = threadIdx.x >> 5;
    int lane = threadIdx.x & 31;
    int tile = blockIdx.x * 8 + wave;
    if (tile >= NTILES) return;
    int m0   = tile * 16;
    int row  = lane & 15;
    int koff = (lane >> 4) * 2;
    int m    = m0 + row;

    const float* aAgg = agg + (size_t)m * DIM;
    const float* aH   = hin + (size_t)m * DIM;
    float sc = 1.0f / fmaxf(cnt[m], 1.0f);

    v8f acc[8];
    v8f zv = {0.f,0.f,0.f,0.f,0.f,0.f,0.f,0.f};
#pragma unroll
    for (int t = 0; t < 8; ++t) acc[t] = zv;

    for (int k = 0; k < DIM; k += 4) {
        v2f fA, fH;
        fA.x = aAgg[k + koff] * sc;  fA.y = aAgg[k + koff + 1] * sc;
        fH.x = aH[k + koff];         fH.y = aH[k + koff + 1];
#pragma unroll
        for (int t = 0; t < 8; ++t) {
            int n = t * 16 + row;
            v2f bL, bR;
            bL.x = sWl[(k + koff) * DIM + n];  bL.y = sWl[(k + koff + 1) * DIM + n];
            bR.x = sWr[(k + koff) * DIM + n];  bR.y = sWr[(k + koff + 1) * DIM + n];
            acc[t] = __builtin_amdgcn_wmma_f32_16x16x4_f32(false, fA, false, bL,
                                                           (short)0, acc[t], false, false);
            acc[t] = __builtin_amdgcn_wmma_f32_16x16x4_f32(false, fH, false, bR,
                                                           (short)0, acc[t], false, false);
        }
    }

    // head: out[m] = sum_n (h2[m][n] + b1[n]) * Wfc[n] + bfc
    int half = lane >> 4;
    float part[8];
#pragma unroll
    for (int v = 0; v < 8; ++v) part[v] = 0.0f;
#pragma unroll
    for (int t = 0; t < 8; ++t) {
        int n = t * 16 + row;
        float w = sWfc[n], bv = sB[n];
#pragma unroll
        for (int v = 0; v < 8; ++v) part[v] += (acc[t][v] + bv) * w;
    }
    // reduce over the 16 lanes (columns) within each lane-half
#pragma unroll
    for (int off = 1; off < 16; off <<= 1) {
#pragma unroll
        for (int v = 0; v < 8; ++v) part[v] += __shfl_xor(part[v], off, 32);
    }
    if (row == 0) {
        float b0 = bfc[0];
#pragma unroll
        for (int v = 0; v < 8; ++v) out[m0 + v + 8 * half] = part[v] + b0;
    }
}

// --------------------------------------------------------------- launcher
extern "C" void kernel_launch(void* const* d_in, const int* in_sizes, int n_in,
                              void* d_out, int out_size, void* d_ws, size_t ws_size,
                              hipStream_t stream) {
    (void)in_sizes; (void)n_in; (void)out_size; (void)ws_size;
    const float* x    = (const float*)d_in[0];
    const int*   ei   = (const int*)d_in[1];       // [2, E]
    const float* Wl0  = (const float*)d_in[2];
    const float* Wr0  = (const float*)d_in[3];
    const float* b0   = (const float*)d_in[4];
    const float* Wl1  = (const float*)d_in[5];
    const float* Wr1  = (const float*)d_in[6];
    const float* b1   = (const float*)d_in[7];
    const float* Wfc  = (const float*)d_in[8];
    const float* bfc  = (const float*)d_in[9];
    float* out = (float*)d_out;

    const int* src = ei;
    const int* dst = ei + NEDGE;

    float* cnt = (float*)d_ws;                          // N
    float* agg = cnt + N_NODES;                         // N*DIM
    float* h   = agg + (size_t)N_NODES * DIM;           // N*DIM

    const size_t smem_bytes = (2 * DIM * DIM + 2 * DIM) * sizeof(float); // ~129 KB

    // ---- layer 0
    {
        int nz = N_NODES * (DIM + 1);                   // cnt + agg contiguous
        zero_f32<<<(nz + 255) / 256, 256, 0, stream>>>(cnt, nz);
        count_edges<<<(NEDGE + 255) / 256, 256, 0, stream>>>(dst, cnt, NEDGE);
        scatter_add<<<(NEDGE + 7) / 8, 256, 0, stream>>>(x, src, dst, agg, NEDGE);
        sage_gemm_relu<<<(NTILES + 7) / 8, 256, smem_bytes, stream>>>(
            agg, cnt, x, Wl0, Wr0, b0, h);
    }
    // ---- layer 1 + head
    {
        int nz = N_NODES * DIM;
        zero_f32<<<(nz + 255) / 256, 256, 0, stream>>>(agg, nz);
        scatter_add<<<(NEDGE + 7) / 8, 256, 0, stream>>>(h, src, dst, agg, NEDGE);
        sage_gemm_head<<<(NTILES + 7) / 8, 256, smem_bytes, stream>>>(
            agg, cnt, h, Wl1, Wr1, b1, Wfc, bfc, out);
    }
}